// VecDGCNN_seg_67508295958612
// MI455X (gfx1250) — compile-verified
//
#include <hip/hip_runtime.h>

typedef float v2f __attribute__((ext_vector_type(2)));
typedef float v8f __attribute__((ext_vector_type(8)));

#define NEG_SLOPE 0.2f
#define EPSV 1e-6f
#define KNN 16
#define NPTS 2048
#define NBATCH 4

// CDNA5 async DMA gather into LDS (ASYNCcnt-tracked).
#define USE_ASYNC_LDS 1

static __device__ inline v8f wmma_f32(v2f a, v2f b, v8f c) {
  // D = A(16x4 f32) * B(4x16 f32) + C(16x16 f32)
  return __builtin_amdgcn_wmma_f32_16x16x4_f32(false, a, false, b, (short)0, c,
                                               false, false);
}

static __device__ inline v8f zero8() {
  v8f z = {0.f, 0.f, 0.f, 0.f, 0.f, 0.f, 0.f, 0.f};
  return z;
}

// Per-lane scattered global->LDS copy (GLOBAL_LOAD_ASYNC_TO_LDS_B32).
static __device__ inline void async_copy_f32(const float* g, float* l) {
#if USE_ASYNC_LDS
  unsigned loff = (unsigned)(size_t)l;
  asm volatile("global_load_async_to_lds_b32 %0, %1, off"
               :
               : "v"(loff), "v"(g)
               : "memory");
#else
  *l = *g;
#endif
}

static __device__ inline void wait_async() {
#if USE_ASYNC_LDS
  asm volatile("s_wait_asynccnt 0x0" ::: "memory");
#endif
}

// ---------------------------------------------------------------------------
// KNN: for each point, top-16 smallest squared distances (includes self)
// ---------------------------------------------------------------------------
__global__ __launch_bounds__(256) void knn_kernel(const float* __restrict__ x,
                                                  int* __restrict__ idx) {
  __shared__ float px[NPTS], py[NPTS], pz[NPTS];
  int b = blockIdx.x;
  const float* xb = x + (size_t)b * 3 * NPTS;
  for (int m = threadIdx.x; m < NPTS; m += 256) {
    px[m] = xb[0 * NPTS + m];
    py[m] = xb[1 * NPTS + m];
    pz[m] = xb[2 * NPTS + m];
  }
  __syncthreads();
  int n = blockIdx.y * 256 + threadIdx.x;
  float cx = px[n], cy = py[n], cz = pz[n];
  float bd[KNN];
  int bi[KNN];
#pragma unroll
  for (int j = 0; j < KNN; ++j) { bd[j] = 3.4e38f; bi[j] = 0; }
  for (int m = 0; m < NPTS; ++m) {
    float dx = px[m] - cx, dy = py[m] - cy, dz = pz[m] - cz;
    float d2 = dx * dx + dy * dy + dz * dz;
    if (d2 < bd[KNN - 1]) {
#pragma unroll
      for (int j = KNN - 1; j > 0; --j) {
        bool shift = d2 < bd[j - 1];
        float nd = shift ? bd[j - 1] : d2;
        int ni = shift ? bi[j - 1] : m;
        if (d2 < bd[j]) { bd[j] = nd; bi[j] = ni; }
      }
      if (d2 < bd[0]) { bd[0] = d2; bi[0] = m; }
    }
  }
  int* op = idx + ((size_t)b * NPTS + n) * KNN;
#pragma unroll
  for (int j = 0; j < KNN; ++j) op[j] = bi[j];
}

// ---------------------------------------------------------------------------
// Fused edge layer: async-gather graph_feature -> p/d GEMMs (f32 WMMA)
// -> vec_lna -> mean over K.  xin: (B,C,3,N), Wf/Wd: (O,2C), xout: (B,O,3,N)
// One wave per point; 4 waves per block.
// LDS panels are channel-fastest: row = (d*16+k), stride CP=C+2 (bank pad),
// so each 2-channel WMMA operand is one aligned ds_load_b64.
// ---------------------------------------------------------------------------
template <int C, int O>
__global__ __launch_bounds__(128) void edge_layer(
    const float* __restrict__ xin, const float* __restrict__ Wf,
    const float* __restrict__ Wd, const int* __restrict__ idx,
    float* __restrict__ xout) {
  constexpr int C2 = 2 * C;
  constexpr int KC = (C2 + 3) / 4;  // K chunks of 4 for 16x16x4 WMMA
  constexpr int CP = C + 2;         // padded channel stride
  __shared__ float diff[4][48 * CP];  // [(d*16+k)*CP + c]
  __shared__ float cent[4][3 * CP];   // [d*CP + c]
  __shared__ int nsh[4][KNN];

  int lane = threadIdx.x & 31;
  int wave = threadIdx.x >> 5;
  int gid = blockIdx.x * 4 + wave;
  int b = gid / NPTS;
  int n = gid % NPTS;
  const float* xb = xin + (size_t)b * C * 3 * NPTS;

  if (lane < KNN) nsh[wave][lane] = idx[((size_t)b * NPTS + n) * KNN + lane];
  __syncthreads();

  // async DMA gather of the 16 neighbors for all (c,d)
  for (int t = lane; t < C * 48; t += 32) {
    int c = t / 48;
    int rem = t % 48;
    int d = rem / 16, k = rem & 15;
    int m = nsh[wave][k];
    async_copy_f32(&xb[((size_t)c * 3 + d) * NPTS + m],
                   &diff[wave][(d * 16 + k) * CP + c]);
  }
  for (int t = lane; t < C * 3; t += 32) {
    int c = t % C, d = t / C;
    cent[wave][d * CP + c] = xb[((size_t)c * 3 + d) * NPTS + n];
  }
  wait_async();
  __syncthreads();
  // raw neighbor -> (neighbor - center), in place
  for (int t = lane; t < C * 48; t += 32) {
    int dk = t / C, c = t % C;
    diff[wave][dk * CP + c] -= cent[wave][(dk >> 4) * CP + c];
  }
  __syncthreads();

  int lm = lane & 15;
  int hi = lane >> 4;
  const float inv16 = 1.0f / 16.0f;

  for (int ro = 0; ro < O / 16; ++ro) {
    v8f accP[3], accD[3];
#pragma unroll
    for (int d = 0; d < 3; ++d) { accP[d] = zero8(); accD[d] = zero8(); }
    for (int kc = 0; kc < KC; ++kc) {
      int c0 = kc * 4 + hi * 2;
      if constexpr (C2 % 4 == 0 && C % 2 == 0) {
        v2f af = *(const v2f*)&Wf[(ro * 16 + lm) * C2 + c0];
        v2f ad = *(const v2f*)&Wd[(ro * 16 + lm) * C2 + c0];
#pragma unroll
        for (int d = 0; d < 3; ++d) {
          v2f bm = (c0 < C)
                       ? *(const v2f*)&diff[wave][(d * 16 + lm) * CP + c0]
                       : *(const v2f*)&cent[wave][d * CP + (c0 - C)];
          accP[d] = wmma_f32(af, bm, accP[d]);
          accD[d] = wmma_f32(ad, bm, accD[d]);
        }
      } else {  // layer 1 (C=1): scalar build with zero padding
        v2f af, ad;
#pragma unroll
        for (int j = 0; j < 2; ++j) {
          int kk = c0 + j;
          af[j] = (kk < C2) ? Wf[(ro * 16 + lm) * C2 + kk] : 0.f;
          ad[j] = (kk < C2) ? Wd[(ro * 16 + lm) * C2 + kk] : 0.f;
        }
#pragma unroll
        for (int d = 0; d < 3; ++d) {
          v2f bm;
#pragma unroll
          for (int j = 0; j < 2; ++j) {
            int kk = c0 + j;
            float bv = 0.f;
            if (kk < C2)
              bv = (kk < C) ? diff[wave][(d * 16 + lm) * CP + kk]
                            : cent[wave][d * CP + (kk - C)];
            bm[j] = bv;
          }
          accP[d] = wmma_f32(af, bm, accP[d]);
          accD[d] = wmma_f32(ad, bm, accD[d]);
        }
      }
    }
    // vec_lna epilogue (lane-local over d), then mean over k in 16-lane halves
#pragma unroll
    for (int i = 0; i < 8; ++i) {
      float p0 = accP[0][i], p1 = accP[1][i], p2 = accP[2][i];
      float q0 = accD[0][i], q1 = accD[1][i], q2 = accD[2][i];
      float dot = p0 * q0 + p1 * q1 + p2 * q2;
      float dsq = q0 * q0 + q1 * q1 + q2 * q2 + EPSV;
      float s = dot * __builtin_amdgcn_rcpf(dsq);
      bool pos = dot >= 0.f;
      float r0 = NEG_SLOPE * p0 + (1.f - NEG_SLOPE) * (pos ? p0 : p0 - s * q0);
      float r1 = NEG_SLOPE * p1 + (1.f - NEG_SLOPE) * (pos ? p1 : p1 - s * q1);
      float r2 = NEG_SLOPE * p2 + (1.f - NEG_SLOPE) * (pos ? p2 : p2 - s * q2);
      r0 += __shfl_xor(r0, 1, 32); r0 += __shfl_xor(r0, 2, 32);
      r0 += __shfl_xor(r0, 4, 32); r0 += __shfl_xor(r0, 8, 32);
      r1 += __shfl_xor(r1, 1, 32); r1 += __shfl_xor(r1, 2, 32);
      r1 += __shfl_xor(r1, 4, 32); r1 += __shfl_xor(r1, 8, 32);
      r2 += __shfl_xor(r2, 1, 32); r2 += __shfl_xor(r2, 2, 32);
      r2 += __shfl_xor(r2, 4, 32); r2 += __shfl_xor(r2, 8, 32);
      if (lm == 0) {
        int o = ro * 16 + i + 8 * hi;
        xout[(((size_t)b * O + o) * 3 + 0) * NPTS + n] = r0 * inv16;
        xout[(((size_t)b * O + o) * 3 + 1) * NPTS + n] = r1 * inv16;
        xout[(((size_t)b * O + o) * 3 + 2) * NPTS + n] = r2 * inv16;
      }
    }
  }
}

// ---------------------------------------------------------------------------
// Combine layer: xc = vec_lna(concat(x1..x4), wcf(128,240), wcd(1,240)).
// One block per 16-point tile; B panel staged in LDS once (async DMA).
// ---------------------------------------------------------------------------
static __device__ inline const float* fetch_cat_ptr(
    const float* __restrict__ x1, const float* __restrict__ x2,
    const float* __restrict__ x3, const float* __restrict__ x4, int b, int c,
    int d, int n) {
  if (c < 16) return &x1[(((size_t)b * 16 + c) * 3 + d) * NPTS + n];
  if (c < 48) return &x2[(((size_t)b * 32 + (c - 16)) * 3 + d) * NPTS + n];
  if (c < 112) return &x3[(((size_t)b * 64 + (c - 48)) * 3 + d) * NPTS + n];
  return &x4[(((size_t)b * 128 + (c - 112)) * 3 + d) * NPTS + n];
}

__global__ __launch_bounds__(128) void cat_layer(
    const float* __restrict__ x1, const float* __restrict__ x2,
    const float* __restrict__ x3, const float* __restrict__ x4,
    const float* __restrict__ wcf, const float* __restrict__ wcd,
    float* __restrict__ xc) {
  constexpr int CIN = 240, CP = CIN + 2;
  __shared__ float panel[48 * CP];  // [(d*16+k)*CP + c]
  __shared__ float dvsh[48];        // d*16 + k
  int tile = blockIdx.x;
  int b = tile / (NPTS / 16);
  int n0 = (tile % (NPTS / 16)) * 16;

  for (int t = threadIdx.x; t < CIN * 48; t += 128) {
    int c = t / 48;
    int rem = t % 48;
    int d = rem / 16, k = rem & 15;
    async_copy_f32(fetch_cat_ptr(x1, x2, x3, x4, b, c, d, n0 + k),
                   &panel[(d * 16 + k) * CP + c]);
  }
  wait_async();
  __syncthreads();
  if (threadIdx.x < 48) {
    float s = 0.f;
    const float* row = &panel[threadIdx.x * CP];
    for (int c = 0; c < CIN; ++c) s += wcd[c] * row[c];
    dvsh[threadIdx.x] = s;
  }
  __syncthreads();

  int lane = threadIdx.x & 31;
  int wave = threadIdx.x >> 5;
  int lm = lane & 15, hi = lane >> 4;
  int n = n0 + lm;
  float dv0 = dvsh[0 + lm], dv1 = dvsh[16 + lm], dv2 = dvsh[32 + lm];
  float rdsq = __builtin_amdgcn_rcpf(dv0 * dv0 + dv1 * dv1 + dv2 * dv2 + EPSV);

  for (int rr = 0; rr < 2; ++rr) {
    int ro = wave * 2 + rr;  // 8 row tiles over 4 waves
    v8f accP[3];
#pragma unroll
    for (int d = 0; d < 3; ++d) accP[d] = zero8();
    for (int kc = 0; kc < CIN / 4; ++kc) {
      int c0 = kc * 4 + hi * 2;
      v2f af = *(const v2f*)&wcf[(ro * 16 + lm) * CIN + c0];
#pragma unroll
      for (int d = 0; d < 3; ++d) {
        v2f bm = *(const v2f*)&panel[(d * 16 + lm) * CP + c0];
        accP[d] = wmma_f32(af, bm, accP[d]);
      }
    }
#pragma unroll
    for (int i = 0; i < 8; ++i) {
      int o = ro * 16 + i + 8 * hi;
      float p0 = accP[0][i], p1 = accP[1][i], p2 = accP[2][i];
      float dot = p0 * dv0 + p1 * dv1 + p2 * dv2;
      float s = dot * rdsq;
      bool pos = dot >= 0.f;
      float r0 = NEG_SLOPE * p0 + (1.f - NEG_SLOPE) * (pos ? p0 : p0 - s * dv0);
      float r1 = NEG_SLOPE * p1 + (1.f - NEG_SLOPE) * (pos ? p1 : p1 - s * dv1);
      float r2 = NEG_SLOPE * p2 + (1.f - NEG_SLOPE) * (pos ? p2 : p2 - s * dv2);
      xc[(((size_t)b * 128 + o) * 3 + 0) * NPTS + n] = r0;
      xc[(((size_t)b * 128 + o) * 3 + 1) * NPTS + n] = r1;
      xc[(((size_t)b * 128 + o) * 3 + 2) * NPTS + n] = r2;
    }
  }
}

// mean over N: xcm[(b*128+c)*3+d]
__global__ __launch_bounds__(256) void mean_n(const float* __restrict__ xc,
                                              float* __restrict__ xcm) {
  int id = blockIdx.x;
  const float* p = xc + (size_t)id * NPTS;
  float s = 0.f;
  for (int n = threadIdx.x; n < NPTS; n += 256) s += p[n];
  s += __shfl_xor(s, 1, 32); s += __shfl_xor(s, 2, 32);
  s += __shfl_xor(s, 4, 32); s += __shfl_xor(s, 8, 32);
  s += __shfl_xor(s, 16, 32);
  __shared__ float red[8];
  int lane = threadIdx.x & 31, w = threadIdx.x >> 5;
  if (lane == 0) red[w] = s;
  __syncthreads();
  if (threadIdx.x == 0) {
    float t = 0.f;
    for (int i = 0; i < 8; ++i) t += red[i];
    xcm[id] = t * (1.0f / NPTS);
  }
}

// ---------------------------------------------------------------------------
// Dense vec_lna with virtual concat input: c<S from src0 (B,S,3,N), else
// src1 (B,CIN-S,3) broadcast over n. One block per 16-point tile, panel in LDS.
// ---------------------------------------------------------------------------
template <int CIN, int COUT>
__global__ __launch_bounds__(128) void dense_vec_lna(
    const float* __restrict__ src0, int S, const float* __restrict__ src1,
    const float* __restrict__ Wf, const float* __restrict__ Wd,
    float* __restrict__ out) {
  constexpr int CP = CIN + 2;
  __shared__ float panel[48 * CP];  // [(d*16+k)*CP + c]
  int tile = blockIdx.x;
  int b = tile / (NPTS / 16);
  int n0 = (tile % (NPTS / 16)) * 16;

  for (int t = threadIdx.x; t < CIN * 48; t += 128) {
    int c = t / 48;
    int rem = t % 48;
    int d = rem / 16, k = rem & 15;
    const float* g = (c < S)
                         ? &src0[(((size_t)b * S + c) * 3 + d) * NPTS + n0 + k]
                         : &src1[((size_t)b * (CIN - S) + (c - S)) * 3 + d];
    async_copy_f32(g, &panel[(d * 16 + k) * CP + c]);
  }
  wait_async();
  __syncthreads();

  int lane = threadIdx.x & 31;
  int wave = threadIdx.x >> 5;
  int lm = lane & 15, hi = lane >> 4;
  int n = n0 + lm;
  constexpr int RPW = COUT / 64;  // row tiles per wave (4 waves)

  for (int rr = 0; rr < RPW; ++rr) {
    int ro = wave * RPW + rr;
    v8f accP[3], accD[3];
#pragma unroll
    for (int d = 0; d < 3; ++d) { accP[d] = zero8(); accD[d] = zero8(); }
    for (int kc = 0; kc < CIN / 4; ++kc) {
      int c0 = kc * 4 + hi * 2;
      v2f af = *(const v2f*)&Wf[(ro * 16 + lm) * CIN + c0];
      v2f ad = *(const v2f*)&Wd[(ro * 16 + lm) * CIN + c0];
#pragma unroll
      for (int d = 0; d < 3; ++d) {
        v2f bm = *(const v2f*)&panel[(d * 16 + lm) * CP + c0];
        accP[d] = wmma_f32(af, bm, accP[d]);
        accD[d] = wmma_f32(ad, bm, accD[d]);
      }
    }
#pragma unroll
    for (int i = 0; i < 8; ++i) {
      int o = ro * 16 + i + 8 * hi;
      float p0 = accP[0][i], p1 = accP[1][i], p2 = accP[2][i];
      float q0 = accD[0][i], q1 = accD[1][i], q2 = accD[2][i];
      float dot = p0 * q0 + p1 * q1 + p2 * q2;
      float dsq = q0 * q0 + q1 * q1 + q2 * q2 + EPSV;
      float s = dot * __builtin_amdgcn_rcpf(dsq);
      bool pos = dot >= 0.f;
      float r0 = NEG_SLOPE * p0 + (1.f - NEG_SLOPE) * (pos ? p0 : p0 - s * q0);
      float r1 = NEG_SLOPE * p1 + (1.f - NEG_SLOPE) * (pos ? p1 : p1 - s * q1);
      float r2 = NEG_SLOPE * p2 + (1.f - NEG_SLOPE) * (pos ? p2 : p2 - s * q2);
      out[(((size_t)b * COUT + o) * 3 + 0) * NPTS + n] = r0;
      out[(((size_t)b * COUT + o) * 3 + 1) * NPTS + n] = r1;
      out[(((size_t)b * COUT + o) * 3 + 2) * NPTS + n] = r2;
    }
  }
}

// inv[b,o,n] = sum_d xcat2[b,o,d,n] * x0[b,o,d,n]
__global__ __launch_bounds__(256) void final_inv(const float* __restrict__ xc,
                                                 const float* __restrict__ xcm,
                                                 const float* __restrict__ x0,
                                                 float* __restrict__ out) {
  int t = blockIdx.x * 256 + threadIdx.x;
  if (t >= NBATCH * 256 * NPTS) return;
  int n = t % NPTS;
  int o = (t / NPTS) % 256;
  int b = t / (NPTS * 256);
  float s = 0.f;
#pragma unroll
  for (int d = 0; d < 3; ++d) {
    float a = (o < 128) ? xc[(((size_t)b * 128 + o) * 3 + d) * NPTS + n]
                        : xcm[((size_t)b * 128 + (o - 128)) * 3 + d];
    float c = x0[(((size_t)b * 256 + o) * 3 + d) * NPTS + n];
    s += a * c;
  }
  out[t] = s;
}

extern "C" void kernel_launch(void* const* d_in, const int* in_sizes, int n_in,
                              void* d_out, int out_size, void* d_ws,
                              size_t ws_size, hipStream_t stream) {
  const float* x = (const float*)d_in[0];
  const float* w1f = (const float*)d_in[1];
  const float* w1d = (const float*)d_in[2];
  const float* w2f = (const float*)d_in[3];
  const float* w2d = (const float*)d_in[4];
  const float* w3f = (const float*)d_in[5];
  const float* w3d = (const float*)d_in[6];
  const float* w4f = (const float*)d_in[7];
  const float* w4d = (const float*)d_in[8];
  const float* wcf = (const float*)d_in[9];
  const float* wcd = (const float*)d_in[10];
  const float* wi1f = (const float*)d_in[11];
  const float* wi1d = (const float*)d_in[12];
  const float* wi2f = (const float*)d_in[13];
  const float* wi2d = (const float*)d_in[14];

  char* ws = (char*)d_ws;
  size_t off = 0;
  int* idx = (int*)(ws + off); off += sizeof(int) * (size_t)NBATCH * NPTS * KNN;
  float* x1 = (float*)(ws + off); off += sizeof(float) * (size_t)NBATCH * 16 * 3 * NPTS;
  float* x2 = (float*)(ws + off); off += sizeof(float) * (size_t)NBATCH * 32 * 3 * NPTS;
  float* x3 = (float*)(ws + off); off += sizeof(float) * (size_t)NBATCH * 64 * 3 * NPTS;
  float* x4 = (float*)(ws + off); off += sizeof(float) * (size_t)NBATCH * 128 * 3 * NPTS;
  float* xc = (float*)(ws + off); off += sizeof(float) * (size_t)NBATCH * 128 * 3 * NPTS;
  float* xcm = (float*)(ws + off); off += sizeof(float) * (size_t)NBATCH * 128 * 3;
  float* y1 = (float*)(ws + off); off += sizeof(float) * (size_t)NBATCH * 128 * 3 * NPTS;
  float* x0 = (float*)(ws + off); off += sizeof(float) * (size_t)NBATCH * 256 * 3 * NPTS;

  knn_kernel<<<dim3(NBATCH, NPTS / 256), 256, 0, stream>>>(x, idx);
  edge_layer<1, 16><<<NBATCH * NPTS / 4, 128, 0, stream>>>(x, w1f, w1d, idx, x1);
  edge_layer<16, 32><<<NBATCH * NPTS / 4, 128, 0, stream>>>(x1, w2f, w2d, idx, x2);
  edge_layer<32, 64><<<NBATCH * NPTS / 4, 128, 0, stream>>>(x2, w3f, w3d, idx, x3);
  edge_layer<64, 128><<<NBATCH * NPTS / 4, 128, 0, stream>>>(x3, w4f, w4d, idx, x4);
  cat_layer<<<NBATCH * (NPTS / 16), 128, 0, stream>>>(x1, x2, x3, x4, wcf, wcd, xc);
  mean_n<<<NBATCH * 128 * 3, 256, 0, stream>>>(xc, xcm);
  dense_vec_lna<256, 128><<<NBATCH * (NPTS / 16), 128, 0, stream>>>(
      xc, 128, xcm, wi1f, wi1d, y1);
  dense_vec_lna<128, 256><<<NBATCH * (NPTS / 16), 128, 0, stream>>>(
      y1, 128, xcm, wi2f, wi2d, x0);
  final_inv<<<(NBATCH * 256 * NPTS + 255) / 256, 256, 0, stream>>>(
      xc, xcm, x0, (float*)d_out);
}